// FrequencyAttention_20452634263751
// MI455X (gfx1250) — compile-verified
//
#include <hip/hip_runtime.h>

typedef __attribute__((ext_vector_type(2))) float v2f;
typedef __attribute__((ext_vector_type(8))) float v8f;
typedef __attribute__((ext_vector_type(4))) int   v4i;

#define CH    64
#define NTOK  102400   // 25*64*64 tokens per (b, channel)
#define BATCH 4

#if defined(__has_builtin)
#if __has_builtin(__builtin_amdgcn_global_load_async_to_lds_b128) && \
    __has_builtin(__builtin_amdgcn_s_wait_asynccnt)
#define FA_USE_ASYNC_LDS 1
#endif
#endif

#ifdef FA_USE_ASYNC_LDS
typedef __attribute__((address_space(1))) v4i glb_v4i;   // global int4
typedef __attribute__((address_space(3))) v4i lds_v4i;   // LDS int4
#endif

// ---------------------------------------------------------------------------
// Kernel 0: zero the covariance accumulators in workspace
// ---------------------------------------------------------------------------
__global__ void fa_zero_kernel(float* __restrict__ p, int n) {
  int i = blockIdx.x * blockDim.x + threadIdx.x;
  if (i < n) p[i] = 0.0f;
}

// ---------------------------------------------------------------------------
// Kernel 1: Cov_b = X_b * X_b^T (64x64 per batch), contraction over N.
// f32 WMMA 16x16x4; 16 waves/block, each wave owns one 16x16 tile of Cov.
// Double-buffered GLOBAL_LOAD_ASYNC_TO_LDS_B128 pipeline (ASYNCcnt) hides the
// global->LDS staging behind WMMA issue. LDS rows padded to 68 floats:
// 272B row pitch keeps B128 writes 16B-aligned and makes the ds_load_b64
// fragment reads bank-conflict-free (stride mod 64 = 4).
// The pipeline is branch-free: every iteration issues the "next" chunk
// (last iteration re-issues the final chunk into the dead buffer) so the
// ASYNCcnt wait is a compile-time constant, as the builtin requires.
// ---------------------------------------------------------------------------
#define K1_CHUNK   64           // tokens per LDS buffer (16 K-steps)
#define K1_STRIDE  68           // padded LDS row stride (floats)
#define K1_CHUNKS  100          // chunks per block -> 6400 tokens/block

__global__ __launch_bounds__(512)
void fa_cov_kernel(const float* __restrict__ x, float* __restrict__ cov) {
  __shared__ float xs[2][CH * K1_STRIDE];  // 2 x 17408 B = 34816 B

  const int b    = blockIdx.y;
  const int tid  = threadIdx.x;
  const int lane = tid & 31;
  const int wid  = tid >> 5;               // 0..15
  const int half = lane >> 4;              // 0: K0/K1, 1: K2/K3
  const int lm16 = lane & 15;
  const int rbase = (wid >> 2) * 16;       // A row tile
  const int cbase = (wid & 3)  * 16;       // B col tile

  const float* xb = x + (size_t)b * CH * NTOK
                      + (size_t)blockIdx.x * (K1_CHUNK * K1_CHUNKS);

  // per-thread staging assignment: two 16B transfers cover 64x64 floats
  const int c0 = tid >> 4;                 // rows 0..31
  const int c1 = 32 + (tid >> 4);          // rows 32..63
  const int q  = (tid & 15) * 4;           // float offset within row

#ifdef FA_USE_ASYNC_LDS
  #define FA_ISSUE(pbuf, tok0)                                                 \
    do {                                                                       \
      __builtin_amdgcn_global_load_async_to_lds_b128(                          \
          (glb_v4i*)(xb + (size_t)c0 * NTOK + (tok0) + q),                     \
          (lds_v4i*)&xs[(pbuf)][c0 * K1_STRIDE + q], 0, 0);                    \
      __builtin_amdgcn_global_load_async_to_lds_b128(                          \
          (glb_v4i*)(xb + (size_t)c1 * NTOK + (tok0) + q),                     \
          (lds_v4i*)&xs[(pbuf)][c1 * K1_STRIDE + q], 0, 0);                    \
    } while (0)

  FA_ISSUE(0, 0);                          // prime the pipeline
#endif

  v8f acc = {};
  for (int ch = 0; ch < K1_CHUNKS; ++ch) {
    const int p = ch & 1;
#ifdef FA_USE_ASYNC_LDS
    // Always prefetch a chunk into the other buffer; the final iteration
    // re-issues the last chunk (never read) to keep the wait constant.
    const int nxt = (ch + 1 < K1_CHUNKS) ? (ch + 1) : ch;
    FA_ISSUE(p ^ 1, nxt * K1_CHUNK);
    __builtin_amdgcn_s_wait_asynccnt(2);   // current buffer has landed
#else
    {
      const int tok0 = ch * K1_CHUNK;
      float4 v0 = *(const float4*)(xb + (size_t)c0 * NTOK + tok0 + q);
      float4 v1 = *(const float4*)(xb + (size_t)c1 * NTOK + tok0 + q);
      *(float4*)&xs[p][c0 * K1_STRIDE + q] = v0;
      *(float4*)&xs[p][c1 * K1_STRIDE + q] = v1;
    }
#endif
    __syncthreads();
    #pragma unroll 4
    for (int t = 0; t < K1_CHUNK; t += 4) {
      v2f a  = *(const v2f*)&xs[p][(rbase + lm16) * K1_STRIDE + t + 2 * half];
      v2f bb = *(const v2f*)&xs[p][(cbase + lm16) * K1_STRIDE + t + 2 * half];
      acc = __builtin_amdgcn_wmma_f32_16x16x4_f32(false, a, false, bb,
                                                  (short)0, acc, false, false);
    }
    __syncthreads();   // readers done before this buffer is overwritten
  }

  // D layout: VGPR i -> row (rbase + i + 8*half), col (cbase + lm16)
  float* covb = cov + b * CH * CH;
  #pragma unroll
  for (int i = 0; i < 8; ++i) {
    int row = rbase + i + 8 * half;
    int col = cbase + lm16;
    atomicAdd(&covb[row * CH + col], acc[i]);
  }
}

// ---------------------------------------------------------------------------
// Kernel 2: per-batch small algebra (all 64x64):
//   Tk = Cov*Wk^T; ||k_e||^2 = Wk[e].Tk[:,e]; ||q_d||^2 = Wq[d].Cov.Wq[d]^T
//   att = softmax( (Wq.Tk) / (||q|| ||k||) );  M = att * Wv
// ---------------------------------------------------------------------------
__global__ __launch_bounds__(256)
void fa_small_kernel(const float* __restrict__ W, const float* __restrict__ cov,
                     float* __restrict__ M) {
  __shared__ float Tk[CH * CH];
  __shared__ float att[CH * CH];
  __shared__ float nrm[2 * CH];

  const int b   = blockIdx.x;
  const int tid = threadIdx.x;
  const float* Wq = W;
  const float* Wk = W + CH * CH;
  const float* Wv = W + 2 * CH * CH;
  const float* C  = cov + b * CH * CH;

  for (int i = tid; i < CH * CH; i += 256) {
    int c = i >> 6, e = i & 63;
    float s = 0.0f;
    for (int cc = 0; cc < CH; ++cc) s += C[c * CH + cc] * Wk[e * CH + cc];
    Tk[i] = s;
  }
  __syncthreads();

  if (tid < CH) {                       // ||k_e||
    int e = tid; float s = 0.0f;
    for (int c = 0; c < CH; ++c) s += Wk[e * CH + c] * Tk[c * CH + e];
    nrm[CH + e] = fmaxf(sqrtf(fmaxf(s, 0.0f)), 1e-12f);
  } else if (tid < 2 * CH) {            // ||q_d||
    int d = tid - CH; float s = 0.0f;
    for (int c = 0; c < CH; ++c) {
      float t = 0.0f;
      for (int cc = 0; cc < CH; ++cc) t += C[c * CH + cc] * Wq[d * CH + cc];
      s += Wq[d * CH + c] * t;
    }
    nrm[d] = fmaxf(sqrtf(fmaxf(s, 0.0f)), 1e-12f);
  }
  __syncthreads();

  for (int i = tid; i < CH * CH; i += 256) {
    int d = i >> 6, e = i & 63;
    float s = 0.0f;
    for (int c = 0; c < CH; ++c) s += Wq[d * CH + c] * Tk[c * CH + e];
    att[i] = s / (nrm[d] * nrm[CH + e]);
  }
  __syncthreads();

  if (tid < CH) {                       // row softmax
    int d = tid;
    float m = -3.0e38f;
    for (int e = 0; e < CH; ++e) m = fmaxf(m, att[d * CH + e]);
    float s = 0.0f;
    for (int e = 0; e < CH; ++e) { float ex = __expf(att[d * CH + e] - m); att[d * CH + e] = ex; s += ex; }
    float inv = 1.0f / s;
    for (int e = 0; e < CH; ++e) att[d * CH + e] *= inv;
  }
  __syncthreads();

  float* Mb = M + b * CH * CH;
  for (int i = tid; i < CH * CH; i += 256) {
    int d = i >> 6, c = i & 63;
    float s = 0.0f;
    for (int e = 0; e < CH; ++e) s += att[d * CH + e] * Wv[e * CH + c];
    Mb[i] = s;
  }
}

// ---------------------------------------------------------------------------
// Kernel 3: out_b = M_b (64x64) * X_b (64xN) via f32 WMMA 16x16x4.
// M fragments hoisted into registers (loop-invariant A); B streams from
// global (x is L2-resident: 105 MB < 192 MB L2, so the 4x row-tile reuse
// never re-touches HBM). D stored as 8 contiguous 64B row segments.
// ---------------------------------------------------------------------------
#define K3_TOKENS 128            // tokens per block (8 tiles of 16)

__global__ __launch_bounds__(128)
void fa_out_kernel(const float* __restrict__ x, const float* __restrict__ M,
                   float* __restrict__ out) {
  const int b    = blockIdx.y;
  const int tid  = threadIdx.x;
  const int lane = tid & 31;
  const int wid  = tid >> 5;               // 0..3
  const int half = lane >> 4;
  const int lm16 = lane & 15;
  const int rbase = wid * 16;

  const float* xb = x   + (size_t)b * CH * NTOK;
  const float* Mb = M   + b * CH * CH;
  float*       ob = out + (size_t)b * CH * NTOK;

  v2f a[16];
  #pragma unroll
  for (int kk = 0; kk < 16; ++kk)
    a[kk] = *(const v2f*)&Mb[(rbase + lm16) * CH + 4 * kk + 2 * half];

  const int t0 = blockIdx.x * K3_TOKENS;
  for (int tt = 0; tt < K3_TOKENS; tt += 16) {
    const int t = t0 + tt + lm16;
    v8f acc = {};
    #pragma unroll
    for (int kk = 0; kk < 16; ++kk) {
      const int c = 4 * kk + 2 * half;
      v2f bb;
      bb.x = xb[(size_t)c       * NTOK + t];
      bb.y = xb[(size_t)(c + 1) * NTOK + t];
      acc = __builtin_amdgcn_wmma_f32_16x16x4_f32(false, a[kk], false, bb,
                                                  (short)0, acc, false, false);
    }
    #pragma unroll
    for (int i = 0; i < 8; ++i)
      ob[(size_t)(rbase + i + 8 * half) * NTOK + t] = acc[i];
  }
}

// ---------------------------------------------------------------------------
extern "C" void kernel_launch(void* const* d_in, const int* in_sizes, int n_in,
                              void* d_out, int out_size, void* d_ws, size_t ws_size,
                              hipStream_t stream) {
  const float* x = (const float*)d_in[0];   // [4,64,25,64,64]
  const float* W = (const float*)d_in[1];   // [192,64]
  float* out = (float*)d_out;               // [4,64,25,64,64]

  float* cov = (float*)d_ws;                // 4*64*64 floats (64 KB)
  float* M   = cov + BATCH * CH * CH;       // 4*64*64 floats (64 KB)

  const int covN = BATCH * CH * CH;
  fa_zero_kernel<<<(covN + 255) / 256, 256, 0, stream>>>(cov, covN);

  // 16 blocks per batch * 100 chunks * 64 tokens = 102400 tokens exactly
  fa_cov_kernel<<<dim3(16, BATCH), 512, 0, stream>>>(x, cov);

  fa_small_kernel<<<BATCH, 256, 0, stream>>>(W, cov, M);

  fa_out_kernel<<<dim3(NTOK / K3_TOKENS, BATCH), 128, 0, stream>>>(x, M, out);
}